// EfficientSlidingWindowMultiheadAttention_90022514524886
// MI455X (gfx1250) — compile-verified
//
#include <hip/hip_runtime.h>

typedef __attribute__((ext_vector_type(2))) float v2f;
typedef __attribute__((ext_vector_type(8))) float v8f;

#define WMMA_F32(a, b, c) \
  __builtin_amdgcn_wmma_f32_16x16x4_f32(false, (a), false, (b), (short)0, (c), false, false)

#define S_LEN   2048
#define D_MODEL 1024
#define TD      3072      // 3*D
#define NHEAD   16
#define HDIM    64
#define NEG_BIG -1.0e30f

#define KB      32        // GEMM k-chunk staged in LDS
#define ASTRIDE 34        // padded LDS row stride (dwords) -> conflict-free b64 reads

static __device__ __forceinline__ v2f load2(const float* p) { return *(const v2f*)p; }

// ---------------------------------------------------------------------------
// C[M,N] = A[M,K] @ B[K,N] + bias[N]   (row-major, M%64==0, N%256==0, K%KB==0)
// 4 waves / block, block tile 64(m) x 256(n); all waves share m0 -> A is
// staged once per block into LDS with double-buffered async-to-LDS DMA
// (GLOBAL_LOAD_ASYNC_TO_LDS_B128, ASYNCcnt), overlapping the next chunk's
// copy with the current chunk's WMMAs. B has no cross-wave reuse -> direct.
// Fragment layouts per CDNA5 ISA 7.12.2 (f32 16x16x4):
//   A: lane(half,idx) vgpr j  <-> A[M=idx][K=k0+2*half+j]
//   B: lane(half,idx) vgpr j  <-> B[K=k0+2*half+j][N=idx]
//   C: lane(half,idx) vgpr jv <-> C[M=jv+8*half][N=idx]
// ---------------------------------------------------------------------------
__global__ __launch_bounds__(128) void gemm_bias_kernel(
    const float* __restrict__ A, const float* __restrict__ B,
    const float* __restrict__ bias, float* __restrict__ C,
    int M, int N, int K)
{
  const int tid  = threadIdx.x;
  const int wave = tid >> 5;
  const int lane = tid & 31;
  const int half = lane >> 4;
  const int idx  = lane & 15;
  const int m0 = blockIdx.x * 64;
  const int n0 = blockIdx.y * 256 + wave * 64;

  __shared__ float As[2][64 * ASTRIDE + 2];   // 2 x ~8.5KB double buffer

  const v8f vzero = {0.f, 0.f, 0.f, 0.f, 0.f, 0.f, 0.f, 0.f};
  v8f acc[4][4];
#pragma unroll
  for (int mt = 0; mt < 4; ++mt)
#pragma unroll
    for (int nt = 0; nt < 4; ++nt) acc[mt][nt] = vzero;

  const int NC = K / KB;

  // Stage one 64 x KB chunk of A into LDS buffer `buf` (async, 4 b128/thread).
  auto issue_chunk = [&](int chunk, int buf) {
    const int kc = chunk * KB;
#pragma unroll
    for (int i = 0; i < 4; ++i) {
      const int q  = tid + 128 * i;       // 0..511
      const int r  = q >> 3;              // row 0..63
      const int cs = (q & 7) * 4;         // col segment (4 floats = b128)
      const float* gp = A + (size_t)(m0 + r) * K + kc + cs;
      const unsigned lds_off = (unsigned)(uintptr_t)(&As[buf][r * ASTRIDE + cs]);
      asm volatile("global_load_async_to_lds_b128 %0, %1, off"
                   :: "v"(lds_off), "v"(gp) : "memory");
    }
  };

  issue_chunk(0, 0);
  for (int c = 0; c < NC; ++c) {
    const int buf = c & 1;
    const bool more = (c + 1 < NC);
    if (more) {
      issue_chunk(c + 1, buf ^ 1);
      asm volatile("s_wait_asynccnt 0x4" ::: "memory");  // chunk c landed, c+1 in flight
    } else {
      asm volatile("s_wait_asynccnt 0x0" ::: "memory");
    }
    __syncthreads();                                      // chunk-c visible to all waves

    const float* Asb = &As[buf][0];
    const int kc = c * KB;
#pragma unroll
    for (int ks = 0; ks < KB; ks += 4) {
      const int kk = ks + 2 * half;
      v2f a[4], b[4];
#pragma unroll
      for (int mt = 0; mt < 4; ++mt)
        a[mt] = *(const v2f*)&Asb[(mt * 16 + idx) * ASTRIDE + kk];
#pragma unroll
      for (int nt = 0; nt < 4; ++nt) {
        const float* bp = B + (size_t)(kc + kk) * N + (n0 + nt * 16 + idx);
        v2f t; t.x = bp[0]; t.y = bp[N];
        b[nt] = t;
      }
#pragma unroll
      for (int mt = 0; mt < 4; ++mt)
#pragma unroll
        for (int nt = 0; nt < 4; ++nt)
          acc[mt][nt] = WMMA_F32(a[mt], b[nt], acc[mt][nt]);
    }
    __syncthreads();   // everyone done reading `buf` before chunk c+2 overwrites it
  }

#pragma unroll
  for (int nt = 0; nt < 4; ++nt) {
    const float bv = bias[n0 + nt * 16 + idx];
#pragma unroll
    for (int mt = 0; mt < 4; ++mt)
#pragma unroll
      for (int jv = 0; jv < 8; ++jv) {
        const int m = m0 + mt * 16 + jv + 8 * half;
        C[(size_t)m * N + (n0 + nt * 16 + idx)] = acc[mt][nt][jv] + bv;
      }
  }
}

// ---------------------------------------------------------------------------
// In-place RoPE on q and k slices of qkv[S, 3D]. One thread per (s, head, pair).
// ---------------------------------------------------------------------------
__global__ __launch_bounds__(256) void rope_kernel(
    float* __restrict__ qkv, const float* __restrict__ cosb, const float* __restrict__ sinb)
{
  const int tid = blockIdx.x * blockDim.x + threadIdx.x;
  const int p = tid & 31;          // pair index 0..31
  const int h = (tid >> 5) & 15;   // head
  const int s = tid >> 9;          // position
  if (s >= S_LEN) return;
  const float c  = cosb[s * 32 + p];
  const float sn = sinb[s * 32 + p];
  float* base = qkv + (size_t)s * TD + h * 192;
  float q0 = base[2 * p], q1 = base[2 * p + 1];
  base[2 * p]     = q0 * c - q1 * sn;
  base[2 * p + 1] = q0 * sn + q1 * c;
  float k0 = base[64 + 2 * p], k1 = base[64 + 2 * p + 1];
  base[64 + 2 * p]     = k0 * c - k1 * sn;
  base[64 + 2 * p + 1] = k0 * sn + k1 * c;
}

// ---------------------------------------------------------------------------
// Sliding-window attention: one wave per (head, 16-query block).
// Scores over absolute key positions js0..js0+143 (9 WMMA tiles, K=64).
// OOB keys read as 0 -> score 0 (matches reference zero-padding).
// Outside-the-129-window entries get -1e30 before softmax.
// PV: P[16x144] @ V[144x64] via WMMA, P re-fragmented through LDS.
// ---------------------------------------------------------------------------
__global__ __launch_bounds__(32) void attn_kernel(
    const float* __restrict__ qkv, float* __restrict__ ctxb)
{
  const int lane = threadIdx.x;
  const int half = lane >> 4;
  const int idx  = lane & 15;
  const int s0 = blockIdx.x * 16;
  const int h  = blockIdx.y;

  const float* Q  = qkv + h * 192;
  const float* Kp = Q + 64;
  const float* Vp = Q + 128;

  __shared__ float Pl[16 * 148];   // padded stride -> conflict-free b64 reads

  v2f qa[16];
#pragma unroll
  for (int t = 0; t < 16; ++t)
    qa[t] = load2(Q + (size_t)(s0 + idx) * TD + 4 * t + 2 * half);

  const int js0 = s0 - 64;

  // ---- scores: 9 tiles of 16 columns -------------------------------------
  v8f sc[9];
#pragma unroll
  for (int ct = 0; ct < 9; ++ct) {
    const int j = js0 + ct * 16 + idx;                 // absolute key position
    const bool jin = (j >= 0) && (j < S_LEN);
    const float* kc = Kp + (long)j * TD;
    v8f acc = {0.f, 0.f, 0.f, 0.f, 0.f, 0.f, 0.f, 0.f};
#pragma unroll
    for (int t = 0; t < 16; ++t) {
      const int k = 4 * t + 2 * half;
      v2f b;
      if (jin) { b.x = kc[k]; b.y = kc[k + 1]; } else { b.x = 0.f; b.y = 0.f; }
      acc = WMMA_F32(qa[t], b, acc);
    }
#pragma unroll
    for (int jv = 0; jv < 8; ++jv) {
      const int m = s0 + jv + 8 * half;                // absolute query row
      float v = acc[jv] * 0.125f;                      // 1/sqrt(64)
      const int dj = j - m;
      if (dj < -64 || dj > 64) v = NEG_BIG;
      acc[jv] = v;
    }
    sc[ct] = acc;
  }

  // ---- row softmax (rows live across the 16 lanes of each half) ----------
  float mx[8], inv[8];
#pragma unroll
  for (int jv = 0; jv < 8; ++jv) {
    float m = NEG_BIG;
#pragma unroll
    for (int ct = 0; ct < 9; ++ct) m = fmaxf(m, sc[ct][jv]);
#pragma unroll
    for (int off = 8; off >= 1; off >>= 1)
      m = fmaxf(m, __shfl_xor(m, off, 32));            // xor<16 stays in the half
    mx[jv] = m;
  }
#pragma unroll
  for (int jv = 0; jv < 8; ++jv) {
    float s = 0.f;
#pragma unroll
    for (int ct = 0; ct < 9; ++ct) {
      const float e = __expf(sc[ct][jv] - mx[jv]);
      sc[ct][jv] = e;
      s += e;
    }
#pragma unroll
    for (int off = 8; off >= 1; off >>= 1)
      s += __shfl_xor(s, off, 32);
    inv[jv] = 1.0f / s;
  }

  // ---- stage P to LDS in plain [row][col] order --------------------------
#pragma unroll
  for (int ct = 0; ct < 9; ++ct)
#pragma unroll
    for (int jv = 0; jv < 8; ++jv)
      Pl[(jv + 8 * half) * 148 + ct * 16 + idx] = sc[ct][jv] * inv[jv];

  __syncthreads();

  // ---- PV: ctx[16x64] = P[16x144] @ V[144x64] ----------------------------
  const v8f vzero = {0.f, 0.f, 0.f, 0.f, 0.f, 0.f, 0.f, 0.f};
  v8f ctx[4];
#pragma unroll
  for (int nt = 0; nt < 4; ++nt) ctx[nt] = vzero;

  for (int kt = 0; kt < 36; ++kt) {
    const int kk = 4 * kt + 2 * half;
    const v2f a = load2(&Pl[idx * 148 + kk]);          // P A-fragment from LDS
    const int j0 = js0 + kk;
    const int j1 = j0 + 1;
    const bool in0 = (j0 >= 0) && (j0 < S_LEN);
    const bool in1 = (j1 >= 0) && (j1 < S_LEN);
    const float* v0 = Vp + (long)j0 * TD;
    const float* v1 = Vp + (long)j1 * TD;
#pragma unroll
    for (int nt = 0; nt < 4; ++nt) {
      const int col = nt * 16 + idx;
      v2f b;
      b.x = in0 ? v0[col] : 0.f;
      b.y = in1 ? v1[col] : 0.f;
      ctx[nt] = WMMA_F32(a, b, ctx[nt]);
    }
  }

#pragma unroll
  for (int nt = 0; nt < 4; ++nt)
#pragma unroll
    for (int jv = 0; jv < 8; ++jv) {
      const int m = s0 + jv + 8 * half;
      ctxb[(size_t)m * D_MODEL + h * HDIM + nt * 16 + idx] = ctx[nt][jv];
    }
}

// ---------------------------------------------------------------------------
extern "C" void kernel_launch(void* const* d_in, const int* in_sizes, int n_in,
                              void* d_out, int out_size, void* d_ws, size_t ws_size,
                              hipStream_t stream)
{
  (void)in_sizes; (void)n_in; (void)out_size; (void)ws_size;
  const float* x    = (const float*)d_in[0];
  const float* Wqkv = (const float*)d_in[1];
  const float* bqkv = (const float*)d_in[2];
  const float* Wout = (const float*)d_in[3];
  const float* bout = (const float*)d_in[4];
  const float* cosb = (const float*)d_in[5];
  const float* sinb = (const float*)d_in[6];
  float* out = (float*)d_out;

  float* qkv  = (float*)d_ws;                       // 2048*3072 f32 = 24 MB
  float* ctxb = qkv + (size_t)S_LEN * TD;           // 2048*1024 f32 =  8 MB

  // 1) qkv = x @ Wqkv + bqkv
  gemm_bias_kernel<<<dim3(S_LEN / 64, TD / 256), 128, 0, stream>>>(
      x, Wqkv, bqkv, qkv, S_LEN, TD, D_MODEL);

  // 2) RoPE in place on q,k
  rope_kernel<<<(S_LEN * NHEAD * 32) / 256, 256, 0, stream>>>(qkv, cosb, sinb);

  // 3) sliding-window attention -> ctx[S, D]
  attn_kernel<<<dim3(S_LEN / 16, NHEAD), 32, 0, stream>>>(qkv, ctxb);

  // 4) out = ctx @ Wout + bout
  gemm_bias_kernel<<<dim3(S_LEN / 64, D_MODEL / 256), 128, 0, stream>>>(
      ctxb, Wout, bout, out, S_LEN, D_MODEL, D_MODEL);
}